// APPNP_model_652835029799
// MI455X (gfx1250) — compile-verified
//
#include <hip/hip_runtime.h>
#include <hip/hip_bf16.h>

// ---------------- problem constants (match reference) ----------------
constexpr int   N_NODES = 65536;
constexpr int   E_EDGES = 1048576;
constexpr int   IN_DIM  = 500;
constexpr int   HID     = 256;
constexpr int   OUTD    = 64;
constexpr float ALPHA   = 0.1f;
constexpr int   KITER   = 10;

constexpr int   LDSTRIDE = HID + 4;   // 260 floats: pad so A-frag LDS reads are bank-conflict-free

typedef __attribute__((ext_vector_type(2))) float v2f;
typedef __attribute__((ext_vector_type(8))) float v8f;

// =====================================================================
// Fused MLP:  h0 = (relu(relu(x@W1+b1)@W2+b2))@W3 + b3
// One wave per 16-row tile; intermediates staged through LDS.
// Uses V_WMMA_F32_16X16X4_F32 (full fp32 matrix path).
// =====================================================================
__global__ void __launch_bounds__(128)
mlp_fused_kernel(const float* __restrict__ x,
                 const float* __restrict__ W1, const float* __restrict__ b1,
                 const float* __restrict__ W2, const float* __restrict__ b2,
                 const float* __restrict__ W3, const float* __restrict__ b3,
                 float* __restrict__ h0out)
{
    extern __shared__ float smem[];

    const int lane  = threadIdx.x & 31;
    const int wave  = threadIdx.x >> 5;
    const int mtile = blockIdx.x * 4 + wave;          // 4096 tiles total, grid exact
    const int m0    = mtile * 16;
    const int r15   = lane & 15;                      // row (A/C col) within tile
    const int khalf = lane >> 4;                      // 0: K pair {0,1}; 1: K pair {2,3}

    float* ldsA = smem + wave * (2 * 16 * LDSTRIDE);  // h1 tile (16 x 256, padded)
    float* ldsB = ldsA + 16 * LDSTRIDE;               // h2 tile

    // ---------------- Layer 1: h1 = relu(x @ W1 + b1), K = 500 ----------------
    v8f acc[16] = {};
    {
        const float* xrow = x + (size_t)(m0 + r15) * IN_DIM;
        for (int ks = 0; ks < IN_DIM / 4; ++ks) {
            const int kc = ks * 4 + khalf * 2;
            v2f a;
            a.x = xrow[kc];
            a.y = xrow[kc + 1];
            const float* w2k  = W1 + (size_t)kc * HID;
            #pragma unroll
            for (int t = 0; t < 16; ++t) {
                const int col = t * 16 + r15;
                v2f b;
                b.x = w2k[col];
                b.y = w2k[HID + col];
                acc[t] = __builtin_amdgcn_wmma_f32_16x16x4_f32(
                    false, a, false, b, (short)0, acc[t], false, false);
            }
        }
        #pragma unroll
        for (int t = 0; t < 16; ++t) {
            const int col = t * 16 + r15;
            const float bb = b1[col];
            #pragma unroll
            for (int r = 0; r < 8; ++r) {
                const int row = khalf * 8 + r;
                float v = acc[t][r] + bb;
                ldsA[row * LDSTRIDE + col] = v > 0.f ? v : 0.f;
            }
        }
    }
    __syncthreads();

    // ---------------- Layer 2: h2 = relu(h1 @ W2 + b2), K = 256 ----------------
    #pragma unroll
    for (int t = 0; t < 16; ++t) acc[t] = v8f{};
    {
        const float* arow = ldsA + r15 * LDSTRIDE;
        for (int ks = 0; ks < HID / 4; ++ks) {
            const int kc = ks * 4 + khalf * 2;
            v2f a;
            a.x = arow[kc];
            a.y = arow[kc + 1];
            const float* w2k = W2 + (size_t)kc * HID;
            #pragma unroll
            for (int t = 0; t < 16; ++t) {
                const int col = t * 16 + r15;
                v2f b;
                b.x = w2k[col];
                b.y = w2k[HID + col];
                acc[t] = __builtin_amdgcn_wmma_f32_16x16x4_f32(
                    false, a, false, b, (short)0, acc[t], false, false);
            }
        }
        #pragma unroll
        for (int t = 0; t < 16; ++t) {
            const int col = t * 16 + r15;
            const float bb = b2[col];
            #pragma unroll
            for (int r = 0; r < 8; ++r) {
                const int row = khalf * 8 + r;
                float v = acc[t][r] + bb;
                ldsB[row * LDSTRIDE + col] = v > 0.f ? v : 0.f;
            }
        }
    }
    __syncthreads();

    // ---------------- Layer 3: h3 = h2 @ W3 + b3, K = 256, N = 64 ----------------
    v8f acc3[4] = {};
    {
        const float* arow = ldsB + r15 * LDSTRIDE;
        for (int ks = 0; ks < HID / 4; ++ks) {
            const int kc = ks * 4 + khalf * 2;
            v2f a;
            a.x = arow[kc];
            a.y = arow[kc + 1];
            const float* w3k = W3 + (size_t)kc * OUTD;
            #pragma unroll
            for (int t = 0; t < 4; ++t) {
                const int col = t * 16 + r15;
                v2f b;
                b.x = w3k[col];
                b.y = w3k[OUTD + col];
                acc3[t] = __builtin_amdgcn_wmma_f32_16x16x4_f32(
                    false, a, false, b, (short)0, acc3[t], false, false);
            }
        }
        #pragma unroll
        for (int t = 0; t < 4; ++t) {
            const int col = t * 16 + r15;
            const float bb = b3[col];
            #pragma unroll
            for (int r = 0; r < 8; ++r) {
                const int row = khalf * 8 + r;
                h0out[(size_t)(m0 + row) * OUTD + col] = acc3[t][r] + bb;
            }
        }
    }
}

// =====================================================================
// Propagation init: next = ALPHA * h0   (folds the +alpha*h0 term)
// =====================================================================
__global__ void __launch_bounds__(256)
init_alpha_kernel(const float4* __restrict__ h0, float4* __restrict__ nxt)
{
    const int i = blockIdx.x * 256 + threadIdx.x;   // grid exact: N*OUTD/4 elements
    float4 v = h0[i];
    nxt[i] = make_float4(ALPHA * v.x, ALPHA * v.y, ALPHA * v.z, ALPHA * v.w);
}

// =====================================================================
// Edge scatter: next[row] += (1-ALPHA) * w_e * cur[col]
// 64 consecutive lanes per edge => coalesced 64-float feature vectors;
// h (16.7MB) + edges (12.6MB) are fully L2-resident on MI455X (192MB L2).
// =====================================================================
__global__ void __launch_bounds__(256)
scatter_kernel(const float* __restrict__ cur, float* __restrict__ nxt,
               const int* __restrict__ erow, const int* __restrict__ ecol,
               const float* __restrict__ ew)
{
    const int e = blockIdx.x * 4 + threadIdx.y;     // blockDim = (64,4); grid exact
    const int f = threadIdx.x;
    const int r = erow[e];
    const int c = ecol[e];
    const float w = ew[e] * (1.0f - ALPHA);
    const float v = cur[(size_t)c * OUTD + f] * w;
    unsafeAtomicAdd(&nxt[(size_t)r * OUTD + f], v);  // -> global_atomic_add_f32
}

// =====================================================================
extern "C" void kernel_launch(void* const* d_in, const int* in_sizes, int n_in,
                              void* d_out, int out_size, void* d_ws, size_t ws_size,
                              hipStream_t stream)
{
    const float* x  = (const float*)d_in[0];
    const float* W1 = (const float*)d_in[1];
    const float* b1 = (const float*)d_in[2];
    const float* W2 = (const float*)d_in[3];
    const float* b2 = (const float*)d_in[4];
    const float* W3 = (const float*)d_in[5];
    const float* b3 = (const float*)d_in[6];
    const float* ew = (const float*)d_in[7];
    const int* erow = (const int*)d_in[8];
    const int* ecol = (const int*)d_in[9];
    float* out = (float*)d_out;

    // workspace: h0 | bufA | bufB  (each N*64 f32 = 16.7MB; total ~50MB)
    float* h0   = (float*)d_ws;
    float* bufA = h0 + (size_t)N_NODES * OUTD;
    float* bufB = bufA + (size_t)N_NODES * OUTD;

    // fused MLP: 1024 blocks x 128 threads (4 waves), 133120B dynamic LDS
    const size_t shmem = (size_t)4 * 2 * 16 * LDSTRIDE * sizeof(float);
    mlp_fused_kernel<<<N_NODES / 16 / 4, 128, shmem, stream>>>(
        x, W1, b1, W2, b2, W3, b3, h0);

    const int n4 = N_NODES * OUTD / 4;
    for (int it = 0; it < KITER; ++it) {
        const float* cur = (it == 0) ? h0 : ((it & 1) ? bufA : bufB);
        float* nxt = (it == KITER - 1) ? out : ((it & 1) ? bufB : bufA);
        init_alpha_kernel<<<n4 / 256, 256, 0, stream>>>(
            (const float4*)h0, (float4*)nxt);
        dim3 blk(64, 4);
        scatter_kernel<<<E_EDGES / 4, blk, 0, stream>>>(cur, nxt, erow, ecol, ew);
    }
}